// RSKHVault_87462714016201
// MI455X (gfx1250) — compile-verified
//
#include <hip/hip_runtime.h>
#include <math.h>

// MI455X roofline: 2^24 knots * 16B (+1B usage) = ~284MB stream @ 23.3TB/s ~= 12us.
// Memory-bound single pass. Similarity via v_wmma_f32_16x16x4_f32: quaternion mult
// is linear in each operand, so step1_i = G a_i and (J*c31)_i = H' a_i with constant
// 4x4 G, H'; sim_i = rowsum(D1 .* D2) where D1 = A x G^T, D2 = A x H'^T.

typedef __attribute__((ext_vector_type(2))) float v2f;
typedef __attribute__((ext_vector_type(8))) float v8f;

#define TPB 256
#define NB_MAX 1024
#define WAVES_PER_BLOCK (TPB / 32)

struct Quat { float w, x, y, z; };

__device__ __forceinline__ Quat qmul(Quat a, Quat b) {
  Quat r;
  r.w = a.w*b.w - a.x*b.x - a.y*b.y - a.z*b.z;
  r.x = a.w*b.x + a.x*b.w + a.y*b.z - a.z*b.y;
  r.y = a.w*b.y - a.x*b.z + a.y*b.w + a.z*b.x;
  r.z = a.w*b.z + a.x*b.y - a.y*b.x + a.z*b.w;
  return r;
}
__device__ __forceinline__ Quat qconj(Quat a) { return {a.w, -a.x, -a.y, -a.z}; }
__device__ __forceinline__ Quat qnorm(Quat a) {
  float s = a.w*a.w + a.x*a.x + a.y*a.y + a.z*a.z;
  float rs = sqrtf(s + 1e-12f);
  return {a.w/rs, a.x/rs, a.y/rs, a.z/rs};
}

// jax.lax.top_k order: value desc, ties -> lower index first.
__device__ __forceinline__ bool better(float av, int ai, float bv, int bi) {
  return (av > bv) || (av == bv && ai < bi);
}

__device__ __forceinline__ void top5_insert(float v, int i, float val[5], int idx[5]) {
  if (!better(v, i, val[4], idx[4])) return;
  val[4] = v; idx[4] = i;
#pragma unroll
  for (int j = 3; j >= 0; --j) {
    if (better(val[j+1], idx[j+1], val[j], idx[j])) {
      float tv = val[j]; val[j] = val[j+1]; val[j+1] = tv;
      int   ti = idx[j]; idx[j] = idx[j+1]; idx[j+1] = ti;
    }
  }
}

// Tree-merge per-thread sorted top-5 lists held in LDS; result in slot 0.
__device__ __forceinline__ void block_reduce_top5(float* sVal, int* sIdx, int tid) {
  __syncthreads();
  for (int stride = TPB / 2; stride >= 1; stride >>= 1) {
    if (tid < stride) {
      int a = tid * 5, b = (tid + stride) * 5;
      float tv[5]; int ti[5];
      int ia = 0, ib = 0;
#pragma unroll
      for (int k = 0; k < 5; ++k) {
        float av = sVal[a + ia]; int aix = sIdx[a + ia];
        float bv = sVal[b + ib]; int bix = sIdx[b + ib];
        if (better(av, aix, bv, bix)) { tv[k] = av; ti[k] = aix; ++ia; }
        else                          { tv[k] = bv; ti[k] = bix; ++ib; }
      }
#pragma unroll
      for (int k = 0; k < 5; ++k) { sVal[a + k] = tv[k]; sIdx[a + k] = ti[k]; }
    }
    __syncthreads();
  }
}

// One 16-knot tile: r = this lane's component pair (already in WMMA A layout),
// ub = 8 usage bytes for this half's knot rows.
__device__ __forceinline__ void process_tile(v2f r, unsigned long long ub, int base,
                                             v2f Bg, v2f Bh, int lane,
                                             float val[5], int idx[5]) {
  float p  = r[0]*r[0] + r[1]*r[1];
  float s  = p + __shfl_xor(p, 16);          // |a|^2 (partner half-lane has other pair)
  float inv = 1.0f / sqrtf(s + 1e-12f);
  v2f A = { r[0] * inv, r[1] * inv };

  v8f C = {};
  v8f D1 = __builtin_amdgcn_wmma_f32_16x16x4_f32(false, A, false, Bg, (short)0, C, false, false);
  v8f D2 = __builtin_amdgcn_wmma_f32_16x16x4_f32(false, A, false, Bh, (short)0, C, false, false);

  // sim_row = sum over columns 0..3 of D1*D2 (J folded into Bh; cols >=4 are zero).
  float sims[8];
#pragma unroll
  for (int q = 0; q < 8; ++q) {
    float tq = D1[q] * D2[q];
    tq += __shfl_xor(tq, 1);
    tq += __shfl_xor(tq, 2);
    sims[q] = tq;   // valid in lanes 0..3 (row q) and 16..19 (row q+8)
  }
  float m = sims[0];
#pragma unroll
  for (int q = 1; q < 8; ++q) m = fmaxf(m, sims[q]);

  // Lanes 0 and 16 own rows 0..7 / 8..15. Rejection gate is safe: this lane's
  // candidate indices are monotonically increasing, so an equal-valued later
  // candidate loses the (value desc, index asc) tie-break anyway.
  if (((lane & 15) == 0) && m > val[4]) {
    const int b0 = base + ((lane >> 4) << 3);
#pragma unroll
    for (int q = 0; q < 8; ++q) {
      if (((ub >> (8 * q)) & 0xffULL) != 0ULL)
        top5_insert(sims[q], b0 + q, val, idx);
    }
  }
}

__global__ void __launch_bounds__(TPB)
vault_topk_stage1(const float* __restrict__ x4, const float* __restrict__ ctx4,
                  const float* __restrict__ vault, const unsigned char* __restrict__ usage,
                  int n, float* __restrict__ wsVal, int* __restrict__ wsIdx) {
  const int tid  = threadIdx.x;
  const int lane = tid & 31;
  const int wave = tid >> 5;
  const int gwave = blockIdx.x * WAVES_PER_BLOCK + wave;
  const int totalWaves = gridDim.x * WAVES_PER_BLOCK;

  // Constant quaternions (uniform scalar loads).
  Quat qn  = qnorm({x4[0], x4[1], x4[2], x4[3]});
  Quat qc  = qnorm({ctx4[0], ctx4[1], ctx4[2], ctx4[3]});
  Quat c12 = qmul(qn, qconj(qc));
  Quat g   = qmul(c12, qc);   // step1_i = g * conj(a_i) = G a_i
  Quat cn  = qconj(qn);       // c31_i   = a_i * cn      = H a_i

  // G = L(g)*J rows; H' = J*R(cn) rows (Re(p*q) signs folded into H).
  float4 G0 = make_float4( g.w,   g.x,   g.y,   g.z);
  float4 G1 = make_float4( g.x,  -g.w,   g.z,  -g.y);
  float4 G2 = make_float4( g.y,  -g.z,  -g.w,   g.x);
  float4 G3 = make_float4( g.z,   g.y,  -g.x,  -g.w);
  float4 H0 = make_float4( cn.w, -cn.x, -cn.y, -cn.z);
  float4 H1 = make_float4(-cn.x, -cn.w, -cn.z,  cn.y);
  float4 H2 = make_float4(-cn.y,  cn.z, -cn.w, -cn.x);
  float4 H3 = make_float4(-cn.z, -cn.y,  cn.x, -cn.w);

  const int  Ncol    = lane & 15;
  const bool lowHalf = lane < 16;
  // B (4x16) layout symmetric to A: VGPR0 row K=(low?0:2), VGPR1 row K=(low?1:3),
  // column N = lane&15. B[k][j] = M[j][k] => D[i][j] = (M a_i)[j]; columns >=4 zero.
  v2f Bg = {0.f, 0.f}, Bh = {0.f, 0.f};
  if (Ncol < 4) {
    float4 Gr = (Ncol == 0) ? G0 : (Ncol == 1) ? G1 : (Ncol == 2) ? G2 : G3;
    float4 Hr = (Ncol == 0) ? H0 : (Ncol == 1) ? H1 : (Ncol == 2) ? H2 : H3;
    if (lowHalf) { Bg[0] = Gr.x; Bg[1] = Gr.y; Bh[0] = Hr.x; Bh[1] = Hr.y; }
    else         { Bg[0] = Gr.z; Bg[1] = Gr.w; Bh[0] = Hr.z; Bh[1] = Hr.w; }
  }

  float val[5]; int idx[5];
#pragma unroll
  for (int k = 0; k < 5; ++k) { val[k] = -INFINITY; idx[k] = 0x7fffffff; }

  const int nTiles = n >> 4;
  // WMMA A layout (16x4 f32): lane L<16 holds K=0,1 of row L; lane L>=16 holds K=2,3
  // of row L-16 -> one coalesced b64 load per lane lands directly in A layout.
  const int laneOff  = lowHalf ? (lane * 4) : ((lane - 16) * 4 + 2);
  const int halfByte = lowHalf ? 0 : 8;     // which 8 usage bytes this half owns

  // Unroll 2 tiles/iteration: both 256B vault loads + both 8B usage loads issued
  // before either compute -> ~2x memory-level parallelism per wave.
  for (int t = gwave; t < nTiles; t += 2 * totalWaves) {
    const int t0 = t, t1 = t + totalWaves;
    const bool has1 = t1 < nTiles;          // wave-uniform (WMMA needs EXEC all-1s)
    const int base0 = t0 << 4;
    v2f r0 = *(const v2f*)(vault + (size_t)base0 * 4 + laneOff);
    unsigned long long ub0 = *(const unsigned long long*)(usage + base0 + halfByte);
    v2f r1 = {0.f, 0.f};
    unsigned long long ub1 = 0ULL;
    int base1 = 0;
    if (has1) {
      base1 = t1 << 4;
      r1  = *(const v2f*)(vault + (size_t)base1 * 4 + laneOff);
      ub1 = *(const unsigned long long*)(usage + base1 + halfByte);
    }
    process_tile(r0, ub0, base0, Bg, Bh, lane, val, idx);
    if (has1) process_tile(r1, ub1, base1, Bg, Bh, lane, val, idx);
  }

  // Remainder knots (n % 16): faithful scalar path, block 0 / wave 0.
  if (blockIdx.x == 0 && wave == 0) {
    for (int i = (nTiles << 4) + lane; i < n; i += 32) {
      Quat a = { vault[(size_t)i*4+0], vault[(size_t)i*4+1],
                 vault[(size_t)i*4+2], vault[(size_t)i*4+3] };
      Quat qv  = qnorm(a);
      Quat c23 = qmul(qc, qconj(qv));
      Quat c31 = qmul(qv, cn);
      Quat s1  = qmul(c12, c23);
      float sim = s1.w*c31.w - s1.x*c31.x - s1.y*c31.y - s1.z*c31.z;
      if (usage[i]) top5_insert(sim, i, val, idx);
    }
  }

  __shared__ float sVal[TPB * 5];
  __shared__ int   sIdx[TPB * 5];
#pragma unroll
  for (int k = 0; k < 5; ++k) { sVal[tid*5+k] = val[k]; sIdx[tid*5+k] = idx[k]; }
  block_reduce_top5(sVal, sIdx, tid);
  if (tid == 0) {
#pragma unroll
    for (int k = 0; k < 5; ++k) { wsVal[blockIdx.x*5+k] = sVal[k]; wsIdx[blockIdx.x*5+k] = sIdx[k]; }
  }
}

__global__ void __launch_bounds__(TPB)
vault_topk_stage2(const float* __restrict__ vault, int n, int numCand,
                  const float* __restrict__ wsVal, const int* __restrict__ wsIdx,
                  float* __restrict__ out) {
  const int tid = threadIdx.x;
  float val[5]; int idx[5];
#pragma unroll
  for (int k = 0; k < 5; ++k) { val[k] = -INFINITY; idx[k] = 0x7fffffff; }
  for (int c = tid; c < numCand; c += TPB) top5_insert(wsVal[c], wsIdx[c], val, idx);

  __shared__ float sVal[TPB * 5];
  __shared__ int   sIdx[TPB * 5];
#pragma unroll
  for (int k = 0; k < 5; ++k) { sVal[tid*5+k] = val[k]; sIdx[tid*5+k] = idx[k]; }
  block_reduce_top5(sVal, sIdx, tid);

  if (tid == 0) {
    for (int k = 0; k < 5; ++k) {
      int gi = sIdx[k];
      bool ok = (gi >= 0) && (gi < n);
#pragma unroll
      for (int c = 0; c < 4; ++c)
        out[k*4 + c] = ok ? vault[(size_t)gi*4 + c] : 0.0f;  // raw vault entries
      out[20 + k] = sVal[k];                                  // scores
    }
  }
}

extern "C" void kernel_launch(void* const* d_in, const int* in_sizes, int n_in,
                              void* d_out, int out_size, void* d_ws, size_t ws_size,
                              hipStream_t stream) {
  const float*         x     = (const float*)d_in[0];
  const float*         ctx   = (const float*)d_in[1];
  const float*         vault = (const float*)d_in[2];
  const unsigned char* usage = (const unsigned char*)d_in[3];  // jnp bool -> 1 byte
  const int n = in_sizes[2] / 4;

  // Workspace: per-block top-5 (val,f32 + idx,i32).
  int nb = NB_MAX;
  size_t need = (size_t)nb * 5 * (sizeof(float) + sizeof(int));
  while (nb > 1 && need > ws_size) { nb >>= 1; need = (size_t)nb * 5 * 8; }
  float* wsVal = (float*)d_ws;
  int*   wsIdx = (int*)((char*)d_ws + (size_t)nb * 5 * sizeof(float));

  vault_topk_stage1<<<nb, TPB, 0, stream>>>(x, ctx, vault, usage, n, wsVal, wsIdx);
  vault_topk_stage2<<<1, TPB, 0, stream>>>(vault, n, nb * 5, wsVal, wsIdx, (float*)d_out);
}